// Network_73212012528289
// MI455X (gfx1250) — compile-verified
//
#include <hip/hip_runtime.h>
#include <cstdint>
#include <cstddef>

// Problem constants (from reference): img [8, 256, 112, 112] fp32
#define B_     8
#define C_     256
#define H_     112
#define W_     112
#define HW_    (H_ * W_)          // 12544
#define NSORT  16384              // next pow2 >= HW_
#define CHUNK  32                 // elements per thread held in VGPRs
#define TSORT  512                // NSORT / CHUNK
#define KSEL   192                // int(0.75 * 256)
#define DENOMF 12996.0f           // (H+2)*(W+2), per reference quirk

// LDS layout for the sort array: pad one word every 32 -> stride 33,
// gcd(33,64)=1 => conflict-free per-chunk sweeps. 16384 + 512 = 16896 slots.
#define SRTN   (NSORT + (NSORT / 32))
#define IDX(i) ((i) + ((i) >> 5))

// ---- gfx1250 async global->LDS path (probe-confirmed to lower) ----
#ifndef HAVE_ASYNC_LDS
#if defined(__has_builtin)
#if __has_builtin(__builtin_amdgcn_global_load_async_to_lds_b128)
#define HAVE_ASYNC_LDS 1
#endif
#endif
#endif
#ifndef HAVE_ASYNC_LDS
#define HAVE_ASYNC_LDS 0
#endif

typedef int v4i_ __attribute__((vector_size(16)));
typedef __attribute__((address_space(1))) v4i_ gv4_t;  // global int4
typedef __attribute__((address_space(3))) v4i_ lv4_t;  // LDS int4

__device__ __forceinline__ void wait_async_zero() {
#if defined(__has_builtin)
#if __has_builtin(__builtin_amdgcn_s_wait_asynccnt)
    __builtin_amdgcn_s_wait_asynccnt(0);
    return;
#else
    asm volatile("s_wait_asynccnt 0" ::: "memory");
    return;
#endif
#else
    asm volatile("s_wait_asynccnt 0" ::: "memory");
#endif
}

// Intra-chunk bitonic passes for stage k: j = min(k/2,16) .. 1, pure VGPRs.
// MUST inline so r[] stays in registers (constant indices after unroll);
// a non-inlined version forces r[] into scratch (observed last round).
template <int N>
__device__ __forceinline__ void intra_sort(float (&r)[N], const int base, const int k) {
    #pragma unroll
    for (int j = 16; j >= 1; j >>= 1) {
        if (j <= (k >> 1)) {                  // pass belongs to this stage
            #pragma unroll
            for (int p = 0; p < N; ++p) {
                if ((p & j) == 0) {
                    const int q = p | j;
                    const bool up = (((base + p) & k) == 0);
                    const float a = r[p], b = r[q];
                    if ((a > b) == up) { r[p] = b; r[q] = a; }
                }
            }
        }
    }
}

// -------------------------------------------------------------------------
// Kernel 1: per (b,c) row -> histogram entropy of ij values.
//   one block per row; dynamic LDS: tile[HW_] | srt[SRTN]
//   Register-blocked bitonic sort: j<=16 passes in VGPRs, j>=32 in LDS.
// -------------------------------------------------------------------------
__global__ __launch_bounds__(TSORT) void entropy_kernel(const float* __restrict__ img,
                                                        float* __restrict__ ent) {
    extern __shared__ float smem[];
    float* tile = smem;          // HW_ floats: the channel image
    float* srt  = smem + HW_;    // SRTN floats: padded sort array
    __shared__ float wsum[TSORT / 32];

    const int row  = blockIdx.x;              // b*C + c
    const int tid  = threadIdx.x;             // 0..511
    const int base = tid * CHUNK;             // this thread's global chunk base
    const float* g = img + (size_t)row * HW_;

    // ---- stage channel image into LDS (async global->LDS, ASYNCcnt) ----
#if HAVE_ASYNC_LDS
    for (int i4 = tid; i4 < HW_ / 4; i4 += TSORT) {
        __builtin_amdgcn_global_load_async_to_lds_b128(
            (gv4_t*)(g + 4 * i4), (lv4_t*)(tile + 4 * i4), 0, 0);
    }
    wait_async_zero();
    __syncthreads();
#else
    {
        const float4* g4 = (const float4*)g;
        float4* t4 = (float4*)tile;
        for (int i4 = tid; i4 < HW_ / 4; i4 += TSORT) t4[i4] = g4[i4];
    }
    __syncthreads();
#endif

    // ---- ij = 100*center + (3x3 zero-padded box sum - center)/8,
    //      computed straight into this thread's register chunk.
    //      HW_ = 392*CHUNK -> the valid/padding split is chunk-uniform.
    float r[CHUNK];
    if (base < HW_) {
        #pragma unroll
        for (int p = 0; p < CHUNK; ++p) {
            const int i = base + p;
            const int y = i / W_;
            const int x = i - y * W_;
            const float c = tile[i];
            float s = 0.0f;
            #pragma unroll
            for (int dy = -1; dy <= 1; ++dy) {
                const int yy = y + dy;
                if (yy < 0 || yy >= H_) continue;
                #pragma unroll
                for (int dx = -1; dx <= 1; ++dx) {
                    const int xx = x + dx;
                    if (xx < 0 || xx >= W_) continue;
                    s += tile[yy * W_ + xx];
                }
            }
            r[p] = fmaf(c, 100.0f, (s - c) * 0.125f);
        }
    } else {
        #pragma unroll
        for (int p = 0; p < CHUNK; ++p) r[p] = __builtin_inff();
    }

    // ---- stages k = 2..32: fully intra-chunk, zero barriers ----
    #pragma unroll
    for (int k = 2; k <= CHUNK; k <<= 1) intra_sort(r, base, k);

    // ---- stages k = 64..NSORT: LDS passes for j>=32, then register tail ----
    for (int k = 2 * CHUNK; k <= NSORT; k <<= 1) {
        #pragma unroll
        for (int p = 0; p < CHUNK; ++p) srt[IDX(base + p)] = r[p];
        __syncthreads();
        for (int j = k >> 1; j >= CHUNK; j >>= 1) {
            for (int i = tid; i < NSORT; i += TSORT) {
                const int ixj = i ^ j;
                if (ixj > i) {
                    const float a = srt[IDX(i)];
                    const float b = srt[IDX(ixj)];
                    const bool up = ((i & k) == 0);
                    if ((a > b) == up) { srt[IDX(i)] = b; srt[IDX(ixj)] = a; }
                }
            }
            __syncthreads();
        }
        #pragma unroll
        for (int p = 0; p < CHUNK; ++p) r[p] = srt[IDX(base + p)];
        intra_sort(r, base, k);
        // next store touches only this thread's own chunk -> no barrier needed
    }

    // publish fully sorted array for the binary-search phase
    #pragma unroll
    for (int p = 0; p < CHUNK; ++p) srt[IDX(base + p)] = r[p];
    __syncthreads();

    // ---- entropy via per-element run counts:
    //      H = (M*log2(D) - sum_e log2(count_e)) / D
    float local = 0.0f;
    if (base < HW_) {                         // padding threads contribute 0
        #pragma unroll 4
        for (int p = 0; p < CHUNK; ++p) {
            const float v = r[p];
            int lo = 0, hi = NSORT;           // lower_bound
            while (lo < hi) { const int m = (lo + hi) >> 1; if (srt[IDX(m)] <  v) lo = m + 1; else hi = m; }
            const int lb = lo;
            lo = 0; hi = NSORT;               // upper_bound
            while (lo < hi) { const int m = (lo + hi) >> 1; if (srt[IDX(m)] <= v) lo = m + 1; else hi = m; }
            local += __log2f((float)(lo - lb));
        }
    }

    // deterministic reduction: wave32 shuffle tree, then fixed-order combine
    #pragma unroll
    for (int off = 16; off > 0; off >>= 1) local += __shfl_xor(local, off, 32);
    const int wid = tid >> 5;
    if ((tid & 31) == 0) wsum[wid] = local;
    __syncthreads();
    if (tid == 0) {
        float tot = 0.0f;
        for (int w = 0; w < TSORT / 32; ++w) tot += wsum[w];
        ent[row] = ((float)HW_ * __log2f(DENOMF) - tot) / DENOMF;
    }
}

// -------------------------------------------------------------------------
// Kernel 2: per batch, top-k (k=192) by entropy; exact lax.top_k ordering
//   (descending value, ties -> lower channel index first).
// -------------------------------------------------------------------------
__global__ __launch_bounds__(C_) void topk_kernel(const float* __restrict__ ent,
                                                  int* __restrict__ idx) {
    __shared__ float e[C_];
    const int b = blockIdx.x;
    const int t = threadIdx.x;       // one thread per channel
    e[t] = ent[b * C_ + t];
    __syncthreads();
    const float v = e[t];
    int rank = 0;
    for (int j = 0; j < C_; ++j) {
        const float ej = e[j];
        rank += (ej > v) || (ej == v && j < t);
    }
    if (rank < KSEL) idx[b * KSEL + rank] = t;
}

// -------------------------------------------------------------------------
// Kernel 3: gather selected channels from the ORIGINAL image.
// -------------------------------------------------------------------------
__global__ __launch_bounds__(256) void gather_kernel(const float* __restrict__ img,
                                                     const int* __restrict__ idx,
                                                     float* __restrict__ out) {
    const int r = blockIdx.x;        // 0..KSEL-1
    const int b = blockIdx.y;        // 0..B_-1
    const int ch = idx[b * KSEL + r];
    const float4* src = (const float4*)(img + ((size_t)b * C_ + ch) * HW_);
    float4*       dst = (float4*)(out + ((size_t)b * KSEL + r) * HW_);
    for (int i = threadIdx.x; i < HW_ / 4; i += blockDim.x) dst[i] = src[i];
}

// -------------------------------------------------------------------------
extern "C" void kernel_launch(void* const* d_in, const int* in_sizes, int n_in,
                              void* d_out, int out_size, void* d_ws, size_t ws_size,
                              hipStream_t stream) {
    (void)in_sizes; (void)n_in; (void)out_size; (void)ws_size;
    const float* img = (const float*)d_in[0];
    float* out = (float*)d_out;

    // workspace: [B*C] entropies, then [B*KSEL] selected indices
    float* ent = (float*)d_ws;
    int*   idx = (int*)((char*)d_ws + (size_t)B_ * C_ * sizeof(float));

    const size_t shmem = (size_t)(HW_ + SRTN) * sizeof(float);  // ~115 KB of 320 KB LDS

    entropy_kernel<<<B_ * C_, TSORT, shmem, stream>>>(img, ent);
    topk_kernel<<<B_, C_, 0, stream>>>(ent, idx);
    gather_kernel<<<dim3(KSEL, B_), 256, 0, stream>>>(img, idx, out);
}